// SaltPepper_38963943309987
// MI455X (gfx1250) — compile-verified
//
#include <hip/hip_runtime.h>

// Salt & pepper noise application:
//   out = (u < 0.01) ? 0.0 : (u < 0.02) ? 1.0 : x
// Pure streaming elementwise op: 12 B of HBM traffic per element, ~604 MB total.
// Memory-bound (floor ~26 us at 23.3 TB/s). Optimized for gfx1250 with:
//  - 128-bit vector loads/stores (global_load_b128 / global_store_b128)
//  - non-temporal cache hints (TH=NT) since the 576 MB working set is
//    touched exactly once and exceeds the 192 MB L2
//  - wave32-friendly 256-thread blocks, 4 independent b128 loads in flight
//    per thread for memory-level parallelism.

typedef float v4f __attribute__((ext_vector_type(4)));

#ifndef SP_HALF
#define SP_HALF 0.01f   // AMOUNT * 0.5
#define SP_FULL 0.02f   // AMOUNT
#endif

__device__ __forceinline__ float sp_one(float x, float u) {
    // u < FULL -> 1.0 (salt), then u < HALF overrides -> 0.0 (pepper)
    float o = (u < SP_FULL) ? 1.0f : x;
    o = (u < SP_HALF) ? 0.0f : o;
    return o;
}

__global__ __launch_bounds__(256) void SaltPepper_vec4_kernel(
    const v4f* __restrict__ img,
    const v4f* __restrict__ u,
    v4f* __restrict__ out,
    long long n4)
{
    constexpr int UNROLL = 4;
    const long long tid   = threadIdx.x;
    const long long bdim  = blockDim.x;
    const long long base  = (long long)blockIdx.x * bdim * UNROLL + tid;

#pragma unroll
    for (int k = 0; k < UNROLL; ++k) {
        const long long i = base + (long long)k * bdim;   // block-strided: coalesced per iter
        if (i < n4) {
            v4f x = __builtin_nontemporal_load(&img[i]);  // global_load_b128 th:NT
            v4f r = __builtin_nontemporal_load(&u[i]);    // global_load_b128 th:NT
            v4f o;
            o.x = sp_one(x.x, r.x);
            o.y = sp_one(x.y, r.y);
            o.z = sp_one(x.z, r.z);
            o.w = sp_one(x.w, r.w);
            __builtin_nontemporal_store(o, &out[i]);      // global_store_b128 th:NT
        }
    }
}

// Scalar tail (element count not divisible by 4). For the reference shapes
// (64*3*512*512) the tail is empty, but keep it for generality.
__global__ __launch_bounds__(256) void SaltPepper_tail_kernel(
    const float* __restrict__ img,
    const float* __restrict__ u,
    float* __restrict__ out,
    long long start, long long n)
{
    long long i = start + (long long)blockIdx.x * blockDim.x + threadIdx.x;
    if (i < n) {
        float x = __builtin_nontemporal_load(&img[i]);
        float r = __builtin_nontemporal_load(&u[i]);
        __builtin_nontemporal_store(sp_one(x, r), &out[i]);
    }
}

extern "C" void kernel_launch(void* const* d_in, const int* in_sizes, int n_in,
                              void* d_out, int out_size, void* d_ws, size_t ws_size,
                              hipStream_t stream) {
    (void)n_in; (void)d_ws; (void)ws_size;

    const float* img = (const float*)d_in[0];   // noised_image (f32)
    const float* u   = (const float*)d_in[1];   // noise_u      (f32)
    float*       out = (float*)d_out;           // f32 output

    const long long n  = (long long)in_sizes[0];
    const long long n4 = n / 4;

    constexpr int THREADS = 256;   // 8 wave32s per block
    constexpr int UNROLL  = 4;

    if (n4 > 0) {
        const long long per_block = (long long)THREADS * UNROLL;     // float4s per block
        const long long blocks    = (n4 + per_block - 1) / per_block;
        SaltPepper_vec4_kernel<<<(unsigned)blocks, THREADS, 0, stream>>>(
            (const v4f*)img, (const v4f*)u, (v4f*)out, n4);
    }

    const long long tail_start = n4 * 4;
    const long long tail       = n - tail_start;
    if (tail > 0) {
        const long long blocks = (tail + THREADS - 1) / THREADS;
        SaltPepper_tail_kernel<<<(unsigned)blocks, THREADS, 0, stream>>>(
            img, u, out, tail_start, n);
    }
}